// InstanceHead_37237366456405
// MI455X (gfx1250) — compile-verified
//
#include <hip/hip_runtime.h>
#include <math.h>

typedef __attribute__((ext_vector_type(2))) float v2f;
typedef __attribute__((ext_vector_type(8))) float v8f;

namespace {

constexpr int SS = 96;                       // spatial_size (fixed by setup_inputs)
constexpr int BB = 2;                        // batch_size
constexpr int PP = SS + 2;                   // 98
constexpr int NCELL = BB * PP * PP * PP;     // 1,882,384 cells (covers +/-1 halo)
constexpr int KTOP = 128;
constexpr int SCAN_CHUNK = 2048;

constexpr int CC = 67;                       // feature dim (LATENT+3)
constexpr int CP = 68;                       // padded to multiple of 4 (WMMA K chunks)
constexpr int KCH = CP / 4;                  // 17 K-chunks for GEMM1
constexpr int DD = 32;                       // descriptor dim
constexpr int OC = KTOP + 1;                 // 129 instance columns
constexpr int NPR = 144;                     // 129 padded to multiple of 16
constexpr int LSTR = 36;                     // LDS row stride (bank-conflict-free, 16B aligned)

__device__ __forceinline__ int cellIdx(int b, int x, int y, int z) {
  return ((b * PP + x + 1) * PP + y + 1) * PP + z + 1;
}

__device__ __forceinline__ unsigned long long packKey(float v, int idx) {
  unsigned u = __float_as_uint(v);
  u = (u & 0x80000000u) ? ~u : (u | 0x80000000u);      // order-preserving float->uint
  return ((unsigned long long)u << 32) | (unsigned long long)(0xFFFFFFFFu - (unsigned)idx);
}

// ---------- revoxelize: keys + occupancy ----------
__global__ void k_keys(const int* __restrict__ coords, const float* __restrict__ offs,
                       int* __restrict__ keyArr, int* __restrict__ occ, int N) {
  int i = blockIdx.x * blockDim.x + threadIdx.x;
  if (i >= N) return;
  int b = coords[i * 4 + 0];
  int x = coords[i * 4 + 1] + (int)offs[i * 3 + 0];
  int y = coords[i * 4 + 2] + (int)offs[i * 3 + 1];
  int z = coords[i * 4 + 3] + (int)offs[i * 3 + 2];
  int key = cellIdx(b, x, y, z);
  keyArr[i] = key;
  occ[key] = 1;
}

// ---------- exclusive scan of occupancy (seg id = rank of key) ----------
__global__ void k_scan1(const int* __restrict__ occ, int* __restrict__ partials, int ncell) {
  __shared__ int sm[256];
  int base = blockIdx.x * SCAN_CHUNK;
  int t = threadIdx.x;
  int s = 0;
  for (int q = 0; q < 8; ++q) {
    int idx = base + t * 8 + q;
    s += (idx < ncell) ? occ[idx] : 0;
  }
  sm[t] = s; __syncthreads();
  for (int st = 128; st > 0; st >>= 1) { if (t < st) sm[t] += sm[t + st]; __syncthreads(); }
  if (t == 0) partials[blockIdx.x] = sm[0];
}

__global__ void k_scan2(int* __restrict__ partials, int nb, int* __restrict__ uCount) {
  int run = 0;
  for (int i = 0; i < nb; ++i) { int v = partials[i]; partials[i] = run; run += v; }
  uCount[0] = run;
}

__global__ void k_scan3(const int* __restrict__ occ, const int* __restrict__ partials,
                        int* __restrict__ scanArr, int ncell) {
  __shared__ int sm[256];
  int base = blockIdx.x * SCAN_CHUNK;
  int t = threadIdx.x;
  int loc[8]; int s = 0;
  for (int q = 0; q < 8; ++q) {
    int idx = base + t * 8 + q;
    int v = (idx < ncell) ? occ[idx] : 0;
    loc[q] = s; s += v;
  }
  sm[t] = s; __syncthreads();
  for (int st = 1; st < 256; st <<= 1) {
    int v = (t >= st) ? sm[t - st] : 0;
    __syncthreads(); sm[t] += v; __syncthreads();
  }
  int off = partials[blockIdx.x] + sm[t] - s;   // exclusive offset for this thread
  for (int q = 0; q < 8; ++q) {
    int idx = base + t * 8 + q;
    if (idx < ncell) scanArr[idx] = off + loc[q];
  }
}

// ---------- per-voxel segment id + segment count / coord sums ----------
__global__ void k_seg(const int* __restrict__ keyArr, const int* __restrict__ scanArr,
                      const int* __restrict__ coords, const float* __restrict__ offs,
                      int* __restrict__ seg, int* __restrict__ counts, int* __restrict__ sumc,
                      int N) {
  int i = blockIdx.x * blockDim.x + threadIdx.x;
  if (i >= N) return;
  int s = scanArr[keyArr[i]];
  seg[i] = s;
  atomicAdd(&counts[s], 1);
  int b = coords[i * 4 + 0];
  int x = coords[i * 4 + 1] + (int)offs[i * 3 + 0];
  int y = coords[i * 4 + 2] + (int)offs[i * 3 + 1];
  int z = coords[i * 4 + 3] + (int)offs[i * 3 + 2];
  atomicAdd(&sumc[s * 4 + 0], b);
  atomicAdd(&sumc[s * 4 + 1], x);
  atomicAdd(&sumc[s * 4 + 2], y);
  atomicAdd(&sumc[s * 4 + 3], z);
}

// ---------- cluster coords (trunc of float mean) + cluster hash grid ----------
__global__ void k_clusgrid(const int* __restrict__ counts, const int* __restrict__ sumc,
                           int* __restrict__ clusCo, int* __restrict__ cgrid,
                           const int* __restrict__ uCount, int N) {
  int s = blockIdx.x * blockDim.x + threadIdx.x;
  if (s >= N || s >= uCount[0]) return;
  float c = (float)counts[s];                 // >= 1 for s < U
  int cb = (int)((float)sumc[s * 4 + 0] / c); // matches jnp float div + int cast
  int cx = (int)((float)sumc[s * 4 + 1] / c);
  int cy = (int)((float)sumc[s * 4 + 2] / c);
  int cz = (int)((float)sumc[s * 4 + 3] / c);
  clusCo[s * 4 + 0] = cb; clusCo[s * 4 + 1] = cx;
  clusCo[s * 4 + 2] = cy; clusCo[s * 4 + 3] = cz;
  atomicMax(&cgrid[cellIdx(cb, cx, cy, cz)], s);
}

// ---------- clus_sum = segment_sum(feats / count_per_voxel) ----------
__global__ void k_clussum(const float* __restrict__ feats, const int* __restrict__ seg,
                          const int* __restrict__ counts, float* __restrict__ clusSum,
                          int N) {
  long long tid = (long long)blockIdx.x * blockDim.x + threadIdx.x;
  if (tid >= (long long)N * CC) return;
  int i = (int)(tid / CC), d = (int)(tid % CC);
  int s = seg[i];
  atomicAdd(&clusSum[(long long)s * CC + d], feats[(long long)i * CC + d] / (float)counts[s]);
}

// ---------- submanifold 3x3x3 avg-pool over cluster sites (stride-padded output) ----------
__global__ void k_pooled(const int* __restrict__ clusCo, const int* __restrict__ cgrid,
                         const float* __restrict__ clusSum, float* __restrict__ pooled,
                         const int* __restrict__ uCount) {
  __shared__ int sj[27];
  int s = blockIdx.x;
  if (s >= uCount[0]) return;
  int t = threadIdx.x;
  int cb = clusCo[s * 4 + 0], cx = clusCo[s * 4 + 1];
  int cy = clusCo[s * 4 + 2], cz = clusCo[s * 4 + 3];
  if (t < 27) {
    int dx = t / 9 - 1, dy = (t / 3) % 3 - 1, dz = t % 3 - 1;
    sj[t] = cgrid[cellIdx(cb, cx + dx, cy + dy, cz + dz)];
  }
  __syncthreads();
  if (t < CC) {
    float acc = 0.f; int cnt = 0;
    for (int q = 0; q < 27; ++q) {            // same offset order as _OFFS
      int j = sj[q];
      if (j >= 0) { acc += clusSum[(long long)j * CC + t]; cnt++; }
    }
    pooled[(long long)s * CP + t] = acc / (float)(cnt > 0 ? cnt : 1);
  } else if (t < CP) {
    pooled[(long long)s * CP + t] = 0.f;      // zero pad column -> unguarded WMMA A loads
  }
}

// ---------- voxel hash grid ----------
__global__ void k_vgrid(const int* __restrict__ coords, int* __restrict__ vgrid, int N) {
  int i = blockIdx.x * blockDim.x + threadIdx.x;
  if (i >= N) return;
  vgrid[cellIdx(coords[i * 4 + 0], coords[i * 4 + 1], coords[i * 4 + 2], coords[i * 4 + 3])] = i;
}

// ---------- 27-tap mixer conv (2 -> 1) + sigmoid ----------
__global__ void k_mixer(const int* __restrict__ coords, const int* __restrict__ vgrid,
                        const float* __restrict__ cscores, const int* __restrict__ counts,
                        const int* __restrict__ seg, const float* __restrict__ mixW,
                        const float* __restrict__ mixB, float* __restrict__ refinedOut,
                        float* __restrict__ score, int N) {
  int i = blockIdx.x * blockDim.x + threadIdx.x;
  if (i >= N) return;
  int b = coords[i * 4 + 0], x = coords[i * 4 + 1];
  int y = coords[i * 4 + 2], z = coords[i * 4 + 3];
  float mix = mixB[0];
  for (int k = 0; k < 27; ++k) {
    int dx = k / 9 - 1, dy = (k / 3) % 3 - 1, dz = k % 3 - 1;
    int j = vgrid[cellIdx(b, x + dx, y + dy, z + dz)];
    if (j >= 0)
      mix += cscores[j] * mixW[k * 2 + 0] + (float)counts[seg[j]] * mixW[k * 2 + 1];
  }
  float r = 1.f / (1.f + expf(-mix));
  refinedOut[i] = r;
  score[i] = r;
}

// ---------- NMS peaks ----------
__global__ void k_peak(const int* __restrict__ coords, const int* __restrict__ vgrid,
                       const float* __restrict__ score, float* __restrict__ peakSc, int N) {
  int i = blockIdx.x * blockDim.x + threadIdx.x;
  if (i >= N) return;
  float s = score[i];
  bool m = s > 0.1f;
  int b = coords[i * 4 + 0], x = coords[i * 4 + 1];
  int y = coords[i * 4 + 2], z = coords[i * 4 + 3];
  float hmax = -INFINITY;
  for (int k = 0; k < 27; ++k) {
    int dx = k / 9 - 1, dy = (k / 3) % 3 - 1, dz = k % 3 - 1;
    int j = vgrid[cellIdx(b, x + dx, y + dy, z + dz)];
    if (j >= 0) {
      float sc = score[j];
      if (sc > 0.1f && sc > hmax) hmax = sc;
    }
  }
  bool peak = m && (hmax == s) && (s > 0.2f);
  peakSc[i] = peak ? s : -1.0f;
}

// ---------- deterministic exact top-K ----------
__global__ void k_tilekeys(const float* __restrict__ peakSc, unsigned long long* __restrict__ tileKeys,
                           int N, int tiles) {
  int t = blockIdx.x * blockDim.x + threadIdx.x;
  if (t >= tiles) return;
  unsigned long long best = 0;
  int e0 = t * 64, e1 = min(e0 + 64, N);
  for (int e = e0; e < e1; ++e) {
    unsigned long long k = packKey(peakSc[e], e);
    if (k > best) best = k;
  }
  tileKeys[t] = best;
}

__global__ void k_topk(float* __restrict__ peakSc, unsigned long long* __restrict__ tileKeys,
                       int tiles, int N, int* __restrict__ topIdx, float* __restrict__ topVal,
                       float* __restrict__ confOut) {
  __shared__ unsigned long long sK[256];
  __shared__ int sT[256];
  int t = threadIdx.x;
  for (int r = 0; r < KTOP; ++r) {
    unsigned long long best = 0; int bt = 0;
    for (int q = t; q < tiles; q += 256) {
      unsigned long long k = tileKeys[q];
      if (k > best) { best = k; bt = q; }
    }
    sK[t] = best; sT[t] = bt;
    __syncthreads();
    for (int st = 128; st > 0; st >>= 1) {
      if (t < st && sK[t + st] > sK[t]) { sK[t] = sK[t + st]; sT[t] = sT[t + st]; }
      __syncthreads();
    }
    if (t == 0) {
      int tile = sT[0];
      int e0 = tile * 64, e1 = min(e0 + 64, N);
      unsigned long long bb = 0; int bi = e0;
      for (int e = e0; e < e1; ++e) {
        unsigned long long k = packKey(peakSc[e], e);
        if (k > bb) { bb = k; bi = e; }
      }
      float v = peakSc[bi];
      topIdx[r] = bi; topVal[r] = v;
      confOut[r] = (v > 0.f) ? v : 0.f;
      peakSc[bi] = -2.0f;                      // remove without replacement
      unsigned long long nb = 0;
      for (int e = e0; e < e1; ++e) {
        unsigned long long k = packKey(peakSc[e], e);
        if (k > nb) nb = k;
      }
      tileKeys[tile] = nb;
    }
    __syncthreads();                            // makes thread-0 global writes visible
  }
}

// ---------- descriptors = [bg; conf_k * relu(peak_feats @ W_desc); zero-pad] ----------
__global__ void k_desc(const float* __restrict__ bg, const float* __restrict__ Wd,
                       const float* __restrict__ pooled, const int* __restrict__ seg,
                       const int* __restrict__ topIdx, const float* __restrict__ topVal,
                       float* __restrict__ descBuf) {
  int n = blockIdx.x; int t = threadIdx.x;
  if (t >= DD) return;
  float out = 0.f;
  if (n == 0) {
    out = bg[t];
  } else if (n <= KTOP) {
    int k = n - 1;
    float v = topVal[k];
    if (v > 0.f) {
      int s = seg[topIdx[k]];
      float acc = 0.f;
      for (int c = 0; c < CC; ++c) acc += pooled[(long long)s * CP + c] * Wd[c * DD + t];
      out = v * fmaxf(acc, 0.f);
    }
  }
  descBuf[n * DD + t] = out;
}

// ---------- WMMA GEMM1: Vdesc[j,:] = relu(pooled[seg[j],:] @ W_desc), j < U ----------
// 4 waves/block, W_desc staged zero-padded into LDS, A gathered once and reused
// across both 16-col tiles, no divergent loads (A rows clamped; pads are zero).
__global__ void __launch_bounds__(128) k_vdesc(const float* __restrict__ pooled,
                                               const int* __restrict__ seg,
                                               const float* __restrict__ Wd,
                                               float* __restrict__ Vdesc,
                                               const int* __restrict__ uCount) {
  __shared__ float sWd[CP * LSTR];
  int tid = threadIdx.x;
  for (int q = tid; q < CP * LSTR; q += 128) {
    int row = q / LSTR, col = q % LSTR;
    sWd[q] = (col < DD && row < CC) ? Wd[row * DD + col] : 0.f;
  }
  __syncthreads();
  int U = uCount[0];
  int wid = tid >> 5, lane = tid & 31;
  int m0 = (blockIdx.x * 4 + wid) * 16;
  if (m0 >= U) return;                           // wave-uniform: EXEC stays all-ones
  int mr = m0 + (lane & 15);
  if (mr >= U) mr = U - 1;                       // clamp: garbage rows are never stored
  const float* Arow = pooled + (long long)seg[mr] * CP;
  __builtin_prefetch(Arow, 0, 3);                // global_prefetch_b8
  int khalf = (lane < 16) ? 0 : 2;
  v2f a[KCH];
#pragma unroll
  for (int c = 0; c < KCH; ++c) {
    int k0 = c * 4 + khalf;
    a[c].x = Arow[k0];
    a[c].y = Arow[k0 + 1];
  }
  int mbase = m0 + ((lane < 16) ? 0 : 8);
#pragma unroll
  for (int nt = 0; nt < 2; ++nt) {
    int ncol = nt * 16 + (lane & 15);
    v8f acc = {};
#pragma unroll
    for (int c = 0; c < KCH; ++c) {
      int k0 = c * 4 + khalf;
      v2f b;
      b.x = sWd[k0 * LSTR + ncol];
      b.y = sWd[(k0 + 1) * LSTR + ncol];
      acc = __builtin_amdgcn_wmma_f32_16x16x4_f32(false, a[c], false, b, (short)0, acc,
                                                  false, false);
    }
#pragma unroll
    for (int r = 0; r < 8; ++r) {
      int rr = mbase + r;
      if (rr < U) Vdesc[(long long)rr * DD + ncol] = fmaxf(acc[r], 0.f);
    }
  }
}

// ---------- WMMA GEMM2: instance[i,:] = Vdesc[seg[i],:] @ descriptors^T ----------
// 4 waves/block share LDS-staged descriptors (bank-conflict-free stride 36);
// per-wave A fragments gathered once, reused across all 9 col tiles.
__global__ void __launch_bounds__(128) k_inst(const float* __restrict__ Vdesc,
                                              const int* __restrict__ seg,
                                              const float* __restrict__ descBuf,
                                              float* __restrict__ instOut, int N) {
  __shared__ float sB[NPR * LSTR];
  int tid = threadIdx.x;
  const float4* src4 = (const float4*)descBuf;
  for (int q = tid; q < NPR * (LSTR / 4); q += 128) {   // 144 rows x 9 float4 slots
    int row = q / (LSTR / 4), c4 = q % (LSTR / 4);
    float4 v = (c4 < DD / 4) ? src4[row * (DD / 4) + c4] : make_float4(0.f, 0.f, 0.f, 0.f);
    *(float4*)&sB[row * LSTR + c4 * 4] = v;
  }
  __syncthreads();
  int wid = tid >> 5, lane = tid & 31;
  int m0 = (blockIdx.x * 4 + wid) * 16;
  if (m0 >= N) return;                           // wave-uniform
  int mr = m0 + (lane & 15);
  if (mr >= N) mr = N - 1;                       // clamp
  const float* Arow = Vdesc + (long long)seg[mr] * DD;
  __builtin_prefetch(Arow, 0, 3);                // global_prefetch_b8
  int khalf = (lane < 16) ? 0 : 2;
  v2f a[DD / 4];
#pragma unroll
  for (int c = 0; c < DD / 4; ++c) {
    int k0 = c * 4 + khalf;
    a[c].x = Arow[k0];
    a[c].y = Arow[k0 + 1];
  }
  int mbase = m0 + ((lane < 16) ? 0 : 8);
#pragma unroll
  for (int nt = 0; nt < NPR / 16; ++nt) {
    int ncol = nt * 16 + (lane & 15);
    v8f acc = {};
#pragma unroll
    for (int c = 0; c < DD / 4; ++c) {
      int k0 = c * 4 + khalf;
      v2f b;
      b.x = sB[ncol * LSTR + k0];                // B[k][n] = descriptors[n][k]
      b.y = sB[ncol * LSTR + k0 + 1];
      acc = __builtin_amdgcn_wmma_f32_16x16x4_f32(false, a[c], false, b, (short)0, acc,
                                                  false, false);
    }
    if (ncol < OC) {
#pragma unroll
      for (int r = 0; r < 8; ++r) {
        int rr = mbase + r;
        if (rr < N) instOut[(long long)rr * OC + ncol] = acc[r];
      }
    }
  }
}

} // namespace

extern "C" void kernel_launch(void* const* d_in, const int* in_sizes, int n_in,
                              void* d_out, int out_size, void* d_ws, size_t ws_size,
                              hipStream_t stream) {
  const int*   coords  = (const int*)d_in[0];
  const float* feats   = (const float*)d_in[1];
  const float* cscores = (const float*)d_in[2];
  const float* offs    = (const float*)d_in[3];
  const float* bg      = (const float*)d_in[4];
  const float* Wd      = (const float*)d_in[5];
  const float* mixW    = (const float*)d_in[6];
  const float* mixB    = (const float*)d_in[7];
  (void)n_in; (void)out_size; (void)ws_size;

  const int N = in_sizes[0] / 4;

  float* refinedOut = (float*)d_out;
  float* confOut    = refinedOut + N;
  float* instOut    = confOut + KTOP;

  // ---- carve workspace ----
  char* w = (char*)d_ws;
  size_t off = 0;
  auto carve = [&](size_t bytes) -> void* {
    size_t a = (off + 255) & ~(size_t)255;
    off = a + bytes;
    return (void*)(w + a);
  };
  int*   occ      = (int*)carve((size_t)NCELL * 4);
  int*   scanArr  = (int*)carve((size_t)NCELL * 4);
  int*   cgrid    = (int*)carve((size_t)NCELL * 4);
  int*   vgrid    = (int*)carve((size_t)NCELL * 4);
  int*   partials = (int*)carve(4096 * 4);
  int*   uCount   = (int*)carve(256);
  int*   keyArr   = (int*)carve((size_t)N * 4);
  int*   seg      = (int*)carve((size_t)N * 4);
  int*   counts   = (int*)carve((size_t)N * 4);
  int*   sumc     = (int*)carve((size_t)N * 16);
  int*   clusCo   = (int*)carve((size_t)N * 16);
  float* clusSum  = (float*)carve((size_t)N * CC * 4);
  float* pooled   = (float*)carve((size_t)N * CP * 4);
  float* score    = (float*)carve((size_t)N * 4);
  float* peakSc   = (float*)carve((size_t)N * 4);
  int tiles = (N + 63) / 64;
  unsigned long long* tileKeys = (unsigned long long*)carve((size_t)tiles * 8);
  int*   topIdx   = (int*)carve(KTOP * 4);
  float* topVal   = (float*)carve(KTOP * 4);
  float* Vdesc    = (float*)carve((size_t)N * DD * 4);
  float* descBuf  = (float*)carve((size_t)NPR * DD * 4);

  // ---- init (graph-capture-safe async memsets) ----
  hipMemsetAsync(occ,     0,    (size_t)NCELL * 4, stream);
  hipMemsetAsync(cgrid,   0xFF, (size_t)NCELL * 4, stream);   // -1
  hipMemsetAsync(vgrid,   0xFF, (size_t)NCELL * 4, stream);   // -1
  hipMemsetAsync(counts,  0,    (size_t)N * 4, stream);
  hipMemsetAsync(sumc,    0,    (size_t)N * 16, stream);
  hipMemsetAsync(clusSum, 0,    (size_t)N * CC * 4, stream);

  const int nb = (NCELL + SCAN_CHUNK - 1) / SCAN_CHUNK;
  const int gN = (N + 255) / 256;
  dim3 b256(256);

  k_keys<<<gN, b256, 0, stream>>>(coords, offs, keyArr, occ, N);
  k_scan1<<<nb, b256, 0, stream>>>(occ, partials, NCELL);
  k_scan2<<<1, 1, 0, stream>>>(partials, nb, uCount);
  k_scan3<<<nb, b256, 0, stream>>>(occ, partials, scanArr, NCELL);
  k_seg<<<gN, b256, 0, stream>>>(keyArr, scanArr, coords, offs, seg, counts, sumc, N);
  k_clusgrid<<<gN, b256, 0, stream>>>(counts, sumc, clusCo, cgrid, uCount, N);

  long long tot = (long long)N * CC;
  int gNC = (int)((tot + 255) / 256);
  k_clussum<<<gNC, b256, 0, stream>>>(feats, seg, counts, clusSum, N);

  k_pooled<<<N, 96, 0, stream>>>(clusCo, cgrid, clusSum, pooled, uCount);

  k_vgrid<<<gN, b256, 0, stream>>>(coords, vgrid, N);
  k_mixer<<<gN, b256, 0, stream>>>(coords, vgrid, cscores, counts, seg, mixW, mixB,
                                   refinedOut, score, N);
  k_peak<<<gN, b256, 0, stream>>>(coords, vgrid, score, peakSc, N);
  k_tilekeys<<<(tiles + 255) / 256, b256, 0, stream>>>(peakSc, tileKeys, N, tiles);
  k_topk<<<1, b256, 0, stream>>>(peakSc, tileKeys, tiles, N, topIdx, topVal, confOut);
  k_desc<<<NPR, 32, 0, stream>>>(bg, Wd, pooled, seg, topIdx, topVal, descBuf);

  const int gM = (N + 63) / 64;                  // 4 m-tiles (waves) per block
  k_vdesc<<<gM, 128, 0, stream>>>(pooled, seg, Wd, Vdesc, uCount);
  k_inst<<<gM, 128, 0, stream>>>(Vdesc, seg, descBuf, instOut, N);
}